// HarmonicSynth_13194139533410
// MI455X (gfx1250) — compile-verified
//
#include <hip/hip_runtime.h>
#include <math.h>

#define NB        4
#define TFRAMES   1000
#define NH        60
#define HOP       192
#define LLEN      192000          // TFRAMES * HOP
#define CHUNK     256
#define NCHUNK    750             // LLEN / CHUNK
#define NCHUNK_PAD 768            // 48 segments of 16
#define NSEG      48
#define C_PH      1.3089969389957472e-4f   // 2*pi/48000
#define AA_LIM    23520.0f                  // 0.49 * 48000

typedef __attribute__((ext_vector_type(2))) float v2f;
typedef __attribute__((ext_vector_type(8))) float v8f;

__device__ __forceinline__ void interp_coeffs(int l, int& i0, int& i1, float& w) {
    // matches F.interpolate(mode='linear', align_corners=False): scale = T/L = 1/HOP
    float pos = ((float)l + 0.5f) * (1.0f / (float)HOP) - 0.5f;
    pos = fmaxf(pos, 0.0f);
    i0 = (int)pos;                         // floor (pos >= 0)
    i0 = min(i0, TFRAMES - 1);
    i1 = min(i0 + 1, TFRAMES - 1);
    w = pos - (float)i0;
}

// ---------------- Phase A: per-chunk sums of interpolated f0 ----------------
__global__ void k_chunksum(const float* __restrict__ f0,
                           float* __restrict__ chunkSums) {
    __shared__ float red[CHUNK];
    const int b = blockIdx.y, c = blockIdx.x, t = threadIdx.x;
    const int l = c * CHUNK + t;
    int i0, i1; float w;
    interp_coeffs(l, i0, i1, w);
    const float* fb = f0 + b * TFRAMES;
    red[t] = fb[i0] * (1.0f - w) + fb[i1] * w;
    __syncthreads();
    for (int s = CHUNK / 2; s > 0; s >>= 1) {
        if (t < s) red[t] += red[t + s];
        __syncthreads();
    }
    if (t == 0) chunkSums[b * NCHUNK_PAD + c] = red[0];
}

// ---------------- Phase B: scan chunk sums via WMMA triangular matmul -------
// Inclusive scan of 16-segments = L(16x16 lower-tri ones) * X(16x16, col = segment),
// realized as 4 chained V_WMMA_F32_16X16X4_F32 (K=4 slices). One wave per tile.
__global__ void k_scan_offsets(const float* __restrict__ chunkSums,
                               float* __restrict__ chunkOffsets) {
    __shared__ float xs[NCHUNK_PAD];
    __shared__ float ss[NCHUNK_PAD];
    __shared__ float segoff[NSEG];
    const int b = blockIdx.x;
    const int t = threadIdx.x;                 // 0..127 (4 waves)

    for (int i = t; i < NCHUNK_PAD; i += 128)
        xs[i] = (i < NCHUNK) ? chunkSums[b * NCHUNK_PAD + i] : 0.0f;
    __syncthreads();

    const int wave = t >> 5;
    const int lane = t & 31;
    if (wave < 3) {                            // wave-uniform: EXEC all ones in wave
        const int tile = wave;                 // 16 segments per tile
        const int mn = lane & 15;              // A: row M ; B: col N ; (same lane id)
        const int hi = lane >> 4;
        v8f acc = {0.f, 0.f, 0.f, 0.f, 0.f, 0.f, 0.f, 0.f};
        #pragma unroll
        for (int k = 0; k < 4; ++k) {
            const int k0 = 4 * k + 2 * hi;     // K index for vgpr0 of this lane half
            const int k1 = k0 + 1;             // K index for vgpr1
            v2f a, bf;
            // A = lower-triangular ones (inclusive): L[m][j] = (j <= m)
            a.x = (k0 <= mn) ? 1.0f : 0.0f;
            a.y = (k1 <= mn) ? 1.0f : 0.0f;
            // B: X[r][c], column c = segment tile*16+c, row r = element-in-segment
            bf.x = xs[(tile * 16 + mn) * 16 + k0];
            bf.y = xs[(tile * 16 + mn) * 16 + k1];
            acc = __builtin_amdgcn_wmma_f32_16x16x4_f32(
                false, a, false, bf, (short)0, acc, false, false);
        }
        // D layout: vgpr v -> M = v + 8*hi, N = lane&15
        #pragma unroll
        for (int vg = 0; vg < 8; ++vg) {
            const int M = vg + 8 * hi;
            ss[(tile * 16 + mn) * 16 + M] = acc[vg];
        }
    }
    __syncthreads();

    if (t == 0) {                              // exclusive scan of 48 segment totals
        float run = 0.0f;
        for (int s = 0; s < NSEG; ++s) { segoff[s] = run; run += ss[s * 16 + 15]; }
    }
    __syncthreads();

    for (int g = t; g < NCHUNK; g += 128) {
        const int seg = g >> 4;
        // exclusive global prefix = segment offset + (inclusive within seg) - self
        chunkOffsets[b * NCHUNK_PAD + g] = segoff[seg] + ss[g] - xs[g];
    }
}

// ---------------- Phase C: in-block scan + Chebyshev harmonic synthesis -----
__global__ void k_synth(const float* __restrict__ f0,
                        const float* __restrict__ amp,
                        const float* __restrict__ hd,
                        const float* __restrict__ chunkOffsets,
                        float* __restrict__ out) {
    __shared__ float sc[CHUNK];
    __shared__ float hrows[4][NH];
    const int b = blockIdx.y, c = blockIdx.x, t = threadIdx.x;
    const int l = c * CHUNK + t;

    int i0, i1; float w;
    interp_coeffs(l, i0, i1, w);
    const float* fb = f0 + b * TFRAMES;
    const float* ab = amp + b * TFRAMES;
    const float f0v  = fb[i0] * (1.0f - w) + fb[i1] * w;
    const float ampv = ab[i0] * (1.0f - w) + ab[i1] * w;

    // frames touched by this chunk: [bi0, bi0+3] (256 samples span < 1.34 frames)
    int bi0, bi1; float bw;
    interp_coeffs(c * CHUNK, bi0, bi1, bw);
    if (t < 4 * NH) {
        const int row = t / NH, col = t - row * NH;
        const int r = min(bi0 + row, TFRAMES - 1);
        hrows[row][col] = hd[(b * TFRAMES + r) * NH + col];
    }

    // inclusive scan of f0_up over the chunk (Hillis-Steele)
    sc[t] = f0v;
    __syncthreads();
    for (int off = 1; off < CHUNK; off <<= 1) {
        const float v = (t >= off) ? sc[t - off] : 0.0f;
        __syncthreads();
        sc[t] += v;
        __syncthreads();
    }

    const float phase = C_PH * (chunkOffsets[b * NCHUNK_PAD + c] + sc[t]);
    float s1, c1;
    sincosf(phase, &s1, &c1);
    const float c2 = 2.0f * c1;

    const int ri0 = i0 - bi0;
    const int ri1 = min(i0 + 1, TFRAMES - 1) - bi0;

    // sin(h*phase) via Chebyshev recurrence; aa-mask folded per harmonic
    float sprev = 0.0f, scur = s1, accum = 0.0f;
    #pragma unroll 6
    for (int h = 1; h <= NH; ++h) {
        const float wh = hrows[ri0][h - 1] * (1.0f - w) + hrows[ri1][h - 1] * w;
        accum += ((float)h * f0v < AA_LIM) ? scur * wh : 0.0f;
        const float snext = fmaf(c2, scur, -sprev);
        sprev = scur; scur = snext;
    }

    const float unv = (f0v > 0.0f) ? 1.0f : 0.0f;
    out[b * LLEN + l] = accum * ampv * unv;
}

extern "C" void kernel_launch(void* const* d_in, const int* in_sizes, int n_in,
                              void* d_out, int out_size, void* d_ws, size_t ws_size,
                              hipStream_t stream) {
    const float* f0  = (const float*)d_in[0];   // (B, T, 1)
    const float* amp = (const float*)d_in[1];   // (B, T, 1)
    const float* hd  = (const float*)d_in[2];   // (B, T, 60)
    float* out = (float*)d_out;                 // (B, L)

    float* chunkSums = (float*)d_ws;                       // NB * 768 floats
    float* chunkOffs = chunkSums + NB * NCHUNK_PAD;        // NB * 768 floats

    dim3 grid(NCHUNK, NB);
    k_chunksum    <<<grid, CHUNK, 0, stream>>>(f0, chunkSums);
    k_scan_offsets<<<NB, 128, 0, stream>>>(chunkSums, chunkOffs);
    k_synth       <<<grid, CHUNK, 0, stream>>>(f0, amp, hd, chunkOffs, out);
}